// GNN_58703613002016
// MI455X (gfx1250) — compile-verified
//
#include <hip/hip_runtime.h>
#include <hip/hip_bf16.h>
#include <math.h>

typedef __attribute__((ext_vector_type(2))) float v2f;
typedef __attribute__((ext_vector_type(8))) float v8f;

#define GNN_NODES  50000
#define GNN_EDGES  800000
#define GNN_FEAT   128
#define GNN_HID    128
#define GNN_GRAPHS 512

// ---------------- utility kernels ----------------

__global__ __launch_bounds__(256) void gnn_zero_f32(float* __restrict__ p, long long n) {
  long long i = (long long)blockIdx.x * blockDim.x + threadIdx.x;
  if (i < n) p[i] = 0.0f;
}

__global__ __launch_bounds__(256) void gnn_edge_degree(const int* __restrict__ dst,
                                                       float* __restrict__ deg, int E) {
  int e = blockIdx.x * blockDim.x + threadIdx.x;
  if (e < E) atomicAdd(&deg[dst[e]], 1.0f);
}

__global__ __launch_bounds__(256) void gnn_inv_degree(float* __restrict__ deg, int n) {
  int i = blockIdx.x * blockDim.x + threadIdx.x;
  if (i < n) deg[i] = 1.0f / fmaxf(deg[i], 1.0f);
}

// scatter-add of gathered 128-dim edge features: aggr[dst] += h[src]
// one thread per (edge, 4-feature group): float4 gather, 4 L2 atomics.
__global__ __launch_bounds__(256) void gnn_scatter_feat(const float* __restrict__ h,
                                                        const int* __restrict__ src,
                                                        const int* __restrict__ dst,
                                                        float* __restrict__ aggr,
                                                        long long total /* E*32 */) {
  long long i = (long long)blockIdx.x * blockDim.x + threadIdx.x;
  if (i >= total) return;
  const int e  = (int)(i >> 5);
  const int f4 = ((int)i & 31) << 2;
  const int s = src[e];
  const int d = dst[e];
  const float4 v = *(const float4*)(h + ((long long)s << 7) + f4);
  float* p = aggr + ((long long)d << 7) + f4;
  atomicAdd(p + 0, v.x);
  atomicAdd(p + 1, v.y);
  atomicAdd(p + 2, v.z);
  atomicAdd(p + 3, v.w);
}

// ---------------- fused dual-GEMM via fp32 WMMA ----------------
// Y[M x N] = (A1 * rowscale) @ W1 + A2 @ W2 + bias
// One wave computes a 32x64 tile (two 16-row fragments share every B fragment
// -> 1 B dword per WMMA instead of 2). K multiple of 4, N multiple of 64.
// M need not be a multiple of 32: A rows are clamped (EXEC stays all-ones for
// WMMA), epilogue stores are row-guarded.
// ISA 7.12.2 fp32 layouts:
//   A (16x4): lanes 0..15 rows, VGPR pair = K+0/1; lanes 16..31 same rows, K+2/3.
//   B (4x16): lane-half selects K pair, N = lane&15.
//   C/D (16x16): vgpr r holds rows r (lanes 0..15) and r+8 (lanes 16..31).
__global__ __launch_bounds__(256) void gnn_gemm_dual(const float* __restrict__ A1,
                                                     const float* __restrict__ rowscale,
                                                     const float* __restrict__ W1,
                                                     const float* __restrict__ A2,
                                                     const float* __restrict__ W2,
                                                     const float* __restrict__ bias,
                                                     float* __restrict__ Y,
                                                     int M, int K, int N) {
  const int lane = threadIdx.x & 31;
  const int wave = threadIdx.x >> 5;
  const int nTiles = N >> 6;                  // 64-col tiles
  const int mTiles = (M + 31) >> 5;           // 32-row tiles
  const int tile = blockIdx.x * 8 + wave;     // 8 waves per block
  if (tile >= mTiles * nTiles) return;        // wave-uniform: EXEC all-ones
  const int nt = tile / mTiles;               // mt-major: block shares B columns
  const int mt = tile - nt * mTiles;
  const int m0 = mt << 5;
  const int n0 = nt << 6;
  const int half = lane >> 4;                 // 0: K+0/1, 1: K+2/3
  const int l16  = lane & 15;
  const int koff = half << 1;

  const int mLo = min(m0 + l16,      M - 1);  // clamp: keep all lanes active
  const int mHi = min(m0 + 16 + l16, M - 1);
  const float sLo = rowscale[mLo];            // fold 1/deg into A1 (mean aggr)
  const float sHi = rowscale[mHi];
  const float* __restrict__ a1lo = A1 + (size_t)mLo * K;
  const float* __restrict__ a1hi = A1 + (size_t)mHi * K;
  const float* __restrict__ a2lo = A2 + (size_t)mLo * K;
  const float* __restrict__ a2hi = A2 + (size_t)mHi * K;

  v8f accLo[4] = {{0.f,0.f,0.f,0.f,0.f,0.f,0.f,0.f},
                  {0.f,0.f,0.f,0.f,0.f,0.f,0.f,0.f},
                  {0.f,0.f,0.f,0.f,0.f,0.f,0.f,0.f},
                  {0.f,0.f,0.f,0.f,0.f,0.f,0.f,0.f}};
  v8f accHi[4] = {{0.f,0.f,0.f,0.f,0.f,0.f,0.f,0.f},
                  {0.f,0.f,0.f,0.f,0.f,0.f,0.f,0.f},
                  {0.f,0.f,0.f,0.f,0.f,0.f,0.f,0.f},
                  {0.f,0.f,0.f,0.f,0.f,0.f,0.f,0.f}};

  for (int k = 0; k < K; k += 4) {
    const int kk = k + koff;
    v2f fA1lo, fA1hi, fA2lo, fA2hi;
    fA1lo.x = a1lo[kk] * sLo;  fA1lo.y = a1lo[kk + 1] * sLo;
    fA1hi.x = a1hi[kk] * sHi;  fA1hi.y = a1hi[kk + 1] * sHi;
    fA2lo.x = a2lo[kk];        fA2lo.y = a2lo[kk + 1];
    fA2hi.x = a2hi[kk];        fA2hi.y = a2hi[kk + 1];
#pragma unroll
    for (int t = 0; t < 4; ++t) {
      const int n = n0 + (t << 4) + l16;
      v2f b1, b2;
      b1.x = W1[(size_t)kk * N + n];
      b1.y = W1[(size_t)(kk + 1) * N + n];
      b2.x = W2[(size_t)kk * N + n];
      b2.y = W2[(size_t)(kk + 1) * N + n];
      accLo[t] = __builtin_amdgcn_wmma_f32_16x16x4_f32(false, fA1lo, false, b1,
                                                       (short)0, accLo[t], false, false);
      accLo[t] = __builtin_amdgcn_wmma_f32_16x16x4_f32(false, fA2lo, false, b2,
                                                       (short)0, accLo[t], false, false);
      accHi[t] = __builtin_amdgcn_wmma_f32_16x16x4_f32(false, fA1hi, false, b1,
                                                       (short)0, accHi[t], false, false);
      accHi[t] = __builtin_amdgcn_wmma_f32_16x16x4_f32(false, fA2hi, false, b2,
                                                       (short)0, accHi[t], false, false);
    }
  }

  // epilogue: add bias, scatter D fragments (row-guarded at the M boundary)
#pragma unroll
  for (int t = 0; t < 4; ++t) {
    const int n = n0 + (t << 4) + l16;
    const float bv = bias[n];
#pragma unroll
    for (int r = 0; r < 8; ++r) {
      const int m = m0 + r + (half << 3);
      if (m < M)      Y[(size_t)m * N + n]        = accLo[t][r] + bv;
      if (m + 16 < M) Y[(size_t)(m + 16) * N + n] = accHi[t][r] + bv;
    }
  }
}

// ---------------- row-wise L2 normalize (+ optional ReLU) ----------------
__global__ __launch_bounds__(256) void gnn_rownorm(float* __restrict__ Y, int M, int N,
                                                   int do_relu) {
  const int row = blockIdx.x * 8 + (threadIdx.x >> 5);
  const int lane = threadIdx.x & 31;
  if (row >= M) return;
  float4* p = (float4*)(Y + (size_t)row * N);
  const int nv = N >> 2;
  float ss = 0.0f;
  for (int c = lane; c < nv; c += 32) {
    float4 v = p[c];
    ss += v.x * v.x + v.y * v.y + v.z * v.z + v.w * v.w;
  }
#pragma unroll
  for (int o = 16; o > 0; o >>= 1) ss += __shfl_xor(ss, o, 32);
  const float scale = 1.0f / fmaxf(sqrtf(ss), 1e-12f);
  for (int c = lane; c < nv; c += 32) {
    float4 v = p[c];
    v.x *= scale; v.y *= scale; v.z *= scale; v.w *= scale;
    if (do_relu) {
      v.x = fmaxf(v.x, 0.0f); v.y = fmaxf(v.y, 0.0f);
      v.z = fmaxf(v.z, 0.0f); v.w = fmaxf(v.w, 0.0f);
    }
    p[c] = v;
  }
}

// ---------------- global mean pool (segment-sum by batch id) ----------------
// one thread per (node, 4-col group) over 256 cols.
__global__ __launch_bounds__(256) void gnn_pool(const float* __restrict__ h2,
                                                const int* __restrict__ batch,
                                                float* __restrict__ gsum,
                                                float* __restrict__ gcnt,
                                                long long total /* nodes*64 */) {
  long long i = (long long)blockIdx.x * blockDim.x + threadIdx.x;
  if (i >= total) return;
  const long long node = i >> 6;
  const int c4 = ((int)i & 63) << 2;
  const int b = batch[node];
  const float4 v = *(const float4*)(h2 + (node << 8) + c4);
  float* p = gsum + ((long long)b << 8) + c4;
  atomicAdd(p + 0, v.x);
  atomicAdd(p + 1, v.y);
  atomicAdd(p + 2, v.z);
  atomicAdd(p + 3, v.w);
  if (c4 == 0) atomicAdd(&gcnt[b], 1.0f);
}

// ---------------- head: sigmoid(pooled @ Wlin + blin), one wave per graph ----
__global__ __launch_bounds__(256) void gnn_head(const float* __restrict__ gsum,
                                                const float* __restrict__ gcnt,
                                                const float* __restrict__ Wlin,
                                                const float* __restrict__ blin,
                                                float* __restrict__ out, int G) {
  const int g = blockIdx.x * 8 + (threadIdx.x >> 5);
  const int lane = threadIdx.x & 31;
  if (g >= G) return;
  const float inv = 1.0f / fmaxf(gcnt[g], 1.0f);
  float dot = 0.0f;
  for (int c = lane; c < 2 * GNN_HID; c += 32)
    dot += gsum[(size_t)g * (2 * GNN_HID) + c] * inv * Wlin[c];
#pragma unroll
  for (int o = 16; o > 0; o >>= 1) dot += __shfl_xor(dot, o, 32);
  if (lane == 0) out[g] = 1.0f / (1.0f + expf(-(dot + blin[0])));
}

// ---------------- launcher ----------------
extern "C" void kernel_launch(void* const* d_in, const int* in_sizes, int n_in,
                              void* d_out, int out_size, void* d_ws, size_t ws_size,
                              hipStream_t stream) {
  const float* x    = (const float*)d_in[0];
  const int*   edge = (const int*)d_in[1];
  const int*   batch= (const int*)d_in[2];
  const float* W1l  = (const float*)d_in[3];
  const float* b1   = (const float*)d_in[4];
  const float* W1r  = (const float*)d_in[5];
  const float* W2l  = (const float*)d_in[6];
  const float* b2   = (const float*)d_in[7];
  const float* W2r  = (const float*)d_in[8];
  const float* Wlin = (const float*)d_in[9];
  const float* blin = (const float*)d_in[10];
  float* out = (float*)d_out;
  float* ws  = (float*)d_ws;

  const int* src = edge;
  const int* dst = edge + GNN_EDGES;

  // workspace layout (floats)
  float* deg   = ws;                                      // 50176 (padded)
  float* aggr  = deg  + 50176;                            // 50000*128
  float* h1    = aggr + (size_t)GNN_NODES * GNN_HID;      // 50000*128
  float* h2    = h1   + (size_t)GNN_NODES * GNN_HID;      // 50000*256
  float* gsum  = h2   + (size_t)GNN_NODES * 2 * GNN_HID;  // 512*256
  float* gcnt  = gsum + (size_t)GNN_GRAPHS * 2 * GNN_HID; // 512

  const long long nAggr  = (long long)GNN_NODES * GNN_HID;
  const long long eWork  = (long long)GNN_EDGES * 32;   // edge x float4-group
  const long long pWork  = (long long)GNN_NODES * 64;   // node x float4-group

  // degree -> 1/deg
  gnn_zero_f32<<<(GNN_NODES + 255) / 256, 256, 0, stream>>>(deg, GNN_NODES);
  gnn_edge_degree<<<(GNN_EDGES + 255) / 256, 256, 0, stream>>>(dst, deg, GNN_EDGES);
  gnn_inv_degree<<<(GNN_NODES + 255) / 256, 256, 0, stream>>>(deg, GNN_NODES);

  // layer 1: aggregate x, dual GEMM (K=128 -> N=128), L2 norm + relu
  gnn_zero_f32<<<(unsigned)((nAggr + 255) / 256), 256, 0, stream>>>(aggr, nAggr);
  gnn_scatter_feat<<<(unsigned)((eWork + 255) / 256), 256, 0, stream>>>(
      x, src, dst, aggr, eWork);
  {
    const int tiles = ((GNN_NODES + 31) / 32) * (GNN_HID / 64);
    gnn_gemm_dual<<<(tiles + 7) / 8, 256, 0, stream>>>(
        aggr, deg, W1l, x, W1r, b1, h1, GNN_NODES, GNN_FEAT, GNN_HID);
  }
  gnn_rownorm<<<(GNN_NODES + 7) / 8, 256, 0, stream>>>(h1, GNN_NODES, GNN_HID, 1);

  // layer 2: aggregate h1, dual GEMM (K=128 -> N=256), L2 norm
  gnn_zero_f32<<<(unsigned)((nAggr + 255) / 256), 256, 0, stream>>>(aggr, nAggr);
  gnn_scatter_feat<<<(unsigned)((eWork + 255) / 256), 256, 0, stream>>>(
      h1, src, dst, aggr, eWork);
  {
    const int tiles = ((GNN_NODES + 31) / 32) * (2 * GNN_HID / 64);
    gnn_gemm_dual<<<(tiles + 7) / 8, 256, 0, stream>>>(
        aggr, deg, W2l, h1, W2r, b2, h2, GNN_NODES, GNN_HID, 2 * GNN_HID);
  }
  gnn_rownorm<<<(GNN_NODES + 7) / 8, 256, 0, stream>>>(h2, GNN_NODES, 2 * GNN_HID, 0);

  // global mean pool + head
  gnn_zero_f32<<<(GNN_GRAPHS * 2 * GNN_HID + 255) / 256, 256, 0, stream>>>(
      gsum, GNN_GRAPHS * 2 * GNN_HID);
  gnn_zero_f32<<<(GNN_GRAPHS + 255) / 256, 256, 0, stream>>>(gcnt, GNN_GRAPHS);
  gnn_pool<<<(unsigned)((pWork + 255) / 256), 256, 0, stream>>>(h2, batch, gsum, gcnt, pWork);
  gnn_head<<<(GNN_GRAPHS + 7) / 8, 256, 0, stream>>>(gsum, gcnt, Wlin, blin, out, GNN_GRAPHS);
}